// YFeatureExtractor_77893526880713
// MI455X (gfx1250) — compile-verified
//
#include <hip/hip_runtime.h>
#include <math.h>

typedef __attribute__((ext_vector_type(2))) float v2f;
typedef __attribute__((ext_vector_type(8))) float v8f;

#define NBASIS      10
#define MAX_RADIUS  10.0f
#define NUM_GRAPHS  1024
#define ZDIM        16
#define FEAT        2560          // 16 * 10 * 16
#define NCOL        160           // 10 * 16  (basis x sh columns)
#define CHUNK       16            // nodes staged in LDS per iteration
#define WAVES_PB    4             // waves (graphs) per block
#define BLOCK_DIM   (WAVES_PB * 32)

// first index i in [0,n) with b[i] >= key  (batch is sorted ascending)
__device__ __forceinline__ int lower_bound_i(const int* __restrict__ b, int n, int key) {
    int lo = 0, hi = n;
    while (lo < hi) {
        int mid = (lo + hi) >> 1;
        if (b[mid] < key) lo = mid + 1; else hi = mid;
    }
    return lo;
}

// column (0..159) -> (basis index, sh index) under per-l-block (b, m) ordering
__device__ __forceinline__ void col_map(int col, int& bi, int& si) {
    if (col < 10)      { bi = col;                 si = 0;                }
    else if (col < 40) { int c = col - 10; bi = c / 3; si = 1 + c % 3;   }
    else if (col < 90) { int c = col - 40; bi = c / 5; si = 4 + c % 5;   }
    else               { int c = col - 90; bi = c / 7; si = 9 + c % 7;   }
}

// (z, col) -> flat feature index with e3nn per-l-block [z, b, m] layout
__device__ __forceinline__ int out_index(int z, int col) {
    if (col < 10)      return 0    + z * 10 + col;
    else if (col < 40) return 160  + z * 30 + (col - 10);
    else if (col < 90) return 640  + z * 50 + (col - 40);
    else               return 1440 + z * 70 + (col - 90);
}

__device__ __forceinline__ float soft_unit_step(float u) {
    return (u > 0.0f) ? __expf(-1.0f / u) : 0.0f;
}

__global__ __launch_bounds__(BLOCK_DIM)
void graph_tp_mean_kernel(const float* __restrict__ x,     // [N,16]
                          const float* __restrict__ pos,   // [N,3]
                          const int*   __restrict__ batch, // [N] sorted
                          float*       __restrict__ out,   // [1024,2560]
                          int n_nodes) {
    __shared__ float wlds[WAVES_PB][CHUNK][NCOL];

    const int lane = threadIdx.x & 31;
    const int wave = threadIdx.x >> 5;
    const int g    = blockIdx.x * WAVES_PB + wave;   // one graph per wave
    if (g >= NUM_GRAPHS) return;

    // graph node range (uniform across the wave)
    const int start = lower_bound_i(batch, n_nodes, g);
    const int end   = lower_bound_i(batch, n_nodes, g + 1);
    const int cnt   = end - start;

    float (*w)[NCOL] = wlds[wave];

    v8f acc[10];
#pragma unroll
    for (int t = 0; t < 10; ++t) acc[t] = (v8f){};

    const int M     = lane & 15;            // A-matrix row / B-matrix col / D col
    const int khalf = (lane >> 4) << 1;     // 0 for lanes 0-15, 2 for lanes 16-31

    for (int c0 = 0; c0 < cnt; c0 += CHUNK) {
        // ---- stage W rows for up to 16 nodes: 2 lanes per node, 80 cols each ----
        {
            const int local = lane >> 1;            // chunk-local node 0..15
            const int sub   = lane & 1;             // half of the 160 columns
            const int node  = c0 + local;
            const float vs  = (node < cnt) ? 1.0f : 0.0f;
            const int gi    = min(start + node, n_nodes - 1);

            const float px = pos[gi * 3 + 0];
            const float py = pos[gi * 3 + 1];
            const float pz = pos[gi * 3 + 2];

            const float r2   = px * px + py * py + pz * pz;
            const float rinv = rsqrtf(r2);
            const float d    = r2 * rinv;
            const float ux = px * rinv, uy = py * rinv, uz = pz * rinv;

            const float s3 = 1.7320508075688772f, s5 = 2.2360679774997896f;
            const float s7 = 2.6457513110645907f, s15 = 3.872983346207417f;
            const float c42  = 1.0801234497346435f;   // sqrt(42)/6
            const float c168 = 1.6201851746019651f;   // sqrt(168)/8

            const float x2 = ux * ux, y2 = uy * uy, z2 = uz * uz;
            const float x2z2 = x2 + z2;

            float sh[16];
            sh[0] = 1.0f;
            sh[1] = s3 * ux; sh[2] = s3 * uy; sh[3] = s3 * uz;
            const float sh20 = s15 * ux * uz;
            const float sh24 = 0.5f * s15 * (z2 - x2);
            sh[4] = sh20;
            sh[5] = s15 * ux * uy;
            sh[6] = s5 * (y2 - 0.5f * x2z2);
            sh[7] = s15 * uy * uz;
            sh[8] = sh24;
            sh[9]  = c42 * (sh20 * uz + sh24 * ux);
            sh[10] = s7 * sh20 * uy;
            sh[11] = c168 * (4.0f * y2 - x2z2) * ux;
            sh[12] = (0.5f * s7) * uy * (2.0f * y2 - 3.0f * x2z2);
            sh[13] = c168 * uz * (4.0f * y2 - x2z2);
            sh[14] = s7 * sh24 * uy;
            sh[15] = c42 * (sh24 * uz - sh20 * ux);

            // smooth_finite radial basis, scaled: C * e^2 * sqrt(NBASIS)
            const float step  = MAX_RADIUS / (NBASIS + 1);
            const float cnorm = 1.14136f * 7.3890560989306495f * 3.1622776601683795f;
            float bas[NBASIS];
#pragma unroll
            for (int j = 0; j < NBASIS; ++j) {
                const float vj   = step * (float)(j + 1);
                const float diff = (d - vj) / step;
                bas[j] = cnorm * soft_unit_step(diff + 1.0f) * soft_unit_step(1.0f - diff);
            }

#pragma unroll
            for (int j = 0; j < NCOL / 2; ++j) {
                const int col = sub * (NCOL / 2) + j;
                int bi, si;
                col_map(col, bi, si);
                w[local][col] = vs * bas[bi] * sh[si];
            }
        }
        // wave-local LDS RAW fence (DS ops are in-order per wave)
        asm volatile("s_wait_dscnt 0" ::: "memory");

        // ---- GEMM over this chunk: K steps of 4 nodes, 10 N-tiles ----
#pragma unroll
        for (int k0 = 0; k0 < CHUNK; k0 += 4) {
            const int na  = c0 + k0 + khalf;                 // graph-local node of a.x
            const float va0 = (na     < cnt) ? 1.0f : 0.0f;
            const float va1 = (na + 1 < cnt) ? 1.0f : 0.0f;
            const int ga0 = min(start + na,     n_nodes - 1);
            const int ga1 = min(start + na + 1, n_nodes - 1);
            v2f a;
            a.x = va0 * x[ga0 * ZDIM + M];
            a.y = va1 * x[ga1 * ZDIM + M];
#pragma unroll
            for (int t = 0; t < 10; ++t) {
                const int col = t * 16 + M;
                v2f b;
                b.x = w[k0 + khalf][col];
                b.y = w[k0 + khalf + 1][col];
                acc[t] = __builtin_amdgcn_wmma_f32_16x16x4_f32(
                    false, a, false, b, (short)0, acc[t], false, false);
            }
        }
        asm volatile("s_wait_dscnt 0" ::: "memory");
    }

    // ---- scatter D tiles to output with segment-mean scaling ----
    const float inv = 1.0f / (float)((cnt > 0) ? cnt : 1);
    const int zbase = (lane >> 4) << 3;     // 0 or 8
#pragma unroll
    for (int t = 0; t < 10; ++t) {
#pragma unroll
        for (int r = 0; r < 8; ++r) {
            const int z   = zbase + r;
            const int col = t * 16 + (lane & 15);
            out[(size_t)g * FEAT + out_index(z, col)] = acc[t][r] * inv;
        }
    }
}

extern "C" void kernel_launch(void* const* d_in, const int* in_sizes, int n_in,
                              void* d_out, int out_size, void* d_ws, size_t ws_size,
                              hipStream_t stream) {
    const float* x     = (const float*)d_in[0];   // [N,16]
    const float* pos   = (const float*)d_in[1];   // [N,3]
    const int*   batch = (const int*)d_in[2];     // [N] sorted graph ids
    float*       out   = (float*)d_out;           // [1024,2560]
    const int n_nodes  = in_sizes[0] / ZDIM;

    dim3 grid(NUM_GRAPHS / WAVES_PB);
    dim3 block(BLOCK_DIM);
    graph_tp_mean_kernel<<<grid, block, 0, stream>>>(x, pos, batch, out, n_nodes);
}